// HypertensorLayer_39006892982497
// MI455X (gfx1250) — compile-verified
//
#include <hip/hip_runtime.h>

typedef __attribute__((ext_vector_type(2))) float v2f;
typedef __attribute__((ext_vector_type(8))) float v8f;

#define B_ROWS 131072
#define DIM    128
#define OUT_N  256

// Each wave computes a 16x64 block of out = (x @ W^T + bias)^2 using
// V_WMMA_F32_16X16X4_F32. 8 waves per workgroup -> 32x256 per WG.
__global__ __launch_bounds__(256, 2)
void hypertensor_linear_sq_kernel(const float* __restrict__ x,
                                  const float* __restrict__ w,
                                  const float* __restrict__ bias,
                                  float* __restrict__ out)
{
    const int lane = threadIdx.x & 31;
    const int wave = threadIdx.x >> 5;
    const int half = lane >> 4;     // lane group: selects K pair (+0 or +2) for A/B,
                                    // and M offset (+0 or +8) for C/D
    const int r    = lane & 15;

    const int wm = wave & 1;        // which 16-row half of the WG's 32 rows
    const int wn = wave >> 1;       // which 64-col group of the 256 cols

    const int row0 = blockIdx.x * 32 + wm * 16;  // wave's first output row
    const int col0 = wn * 64;                    // wave's first output col

    // A layout (16x4 f32): lane L holds x[row0 + L%16][k + 2*(L/16) + {0,1}]
    const float* aptr  = x + (size_t)(row0 + r) * DIM + 2 * half;
    // B layout (4x16 f32), B[k][n] = w[n][k]: lane L holds w[col + L%16][k + 2*(L/16) + {0,1}]
    const float* bptr0 = w + (size_t)(col0 +  0 + r) * DIM + 2 * half;
    const float* bptr1 = w + (size_t)(col0 + 16 + r) * DIM + 2 * half;
    const float* bptr2 = w + (size_t)(col0 + 32 + r) * DIM + 2 * half;
    const float* bptr3 = w + (size_t)(col0 + 48 + r) * DIM + 2 * half;

    v8f c0 = {}, c1 = {}, c2 = {}, c3 = {};

    #pragma unroll 8
    for (int k = 0; k < DIM; k += 4) {
        v2f a  = *(const v2f*)(aptr  + k);
        v2f b0 = *(const v2f*)(bptr0 + k);
        v2f b1 = *(const v2f*)(bptr1 + k);
        v2f b2 = *(const v2f*)(bptr2 + k);
        v2f b3 = *(const v2f*)(bptr3 + k);
        // (neg_a, A, neg_b, B, c_mod, C, reuse_a, reuse_b)
        c0 = __builtin_amdgcn_wmma_f32_16x16x4_f32(false, a, false, b0, (short)0, c0, false, false);
        c1 = __builtin_amdgcn_wmma_f32_16x16x4_f32(false, a, false, b1, (short)0, c1, false, false);
        c2 = __builtin_amdgcn_wmma_f32_16x16x4_f32(false, a, false, b2, (short)0, c2, false, false);
        c3 = __builtin_amdgcn_wmma_f32_16x16x4_f32(false, a, false, b3, (short)0, c3, false, false);
    }

    // bias per N-tile (lanes 0-15 and 16-31 need the same 16 values)
    const float bb0 = bias[col0 +  0 + r];
    const float bb1 = bias[col0 + 16 + r];
    const float bb2 = bias[col0 + 32 + r];
    const float bb3 = bias[col0 + 48 + r];

    // C/D layout: VGPR j, lane L -> row (j + 8*(L/16)), col (L%16) of the 16x16 tile
    #pragma unroll
    for (int j = 0; j < 8; ++j) {
        const size_t rowoff = (size_t)(row0 + j + 8 * half) * OUT_N;
        float z;
        z = c0[j] + bb0; out[rowoff + col0 +  0 + r] = z * z;
        z = c1[j] + bb1; out[rowoff + col0 + 16 + r] = z * z;
        z = c2[j] + bb2; out[rowoff + col0 + 32 + r] = z * z;
        z = c3[j] + bb3; out[rowoff + col0 + 48 + r] = z * z;
    }
}

extern "C" void kernel_launch(void* const* d_in, const int* in_sizes, int n_in,
                              void* d_out, int out_size, void* d_ws, size_t ws_size,
                              hipStream_t stream) {
    (void)in_sizes; (void)n_in; (void)d_ws; (void)ws_size; (void)out_size;
    const float* x    = (const float*)d_in[0];
    const float* w    = (const float*)d_in[1];
    const float* bias = (const float*)d_in[2];
    float* out        = (float*)d_out;

    dim3 grid(B_ROWS / 32);   // 4096 workgroups, each covering 32 rows x 256 cols
    dim3 block(256);          // 8 waves (wave32)
    hipLaunchKernelGGL(hypertensor_linear_sq_kernel, grid, block, 0, stream,
                       x, w, bias, out);
}